// LeakyPlain_11287174053915
// MI455X (gfx1250) — compile-verified
//
#include <hip/hip_runtime.h>

// Leaky integrate-and-fire scan, T=16 timesteps, time = outer (16 MB) stride.
//   u_t = 0.5*u + x_t ; s_t = (u_t > 1) ; u_t -= s_t
// HBM-bound: 256 MB in + 256 MB out -> ~23 us floor at 23.3 TB/s.
//
// gfx1250 design: depth-4 per-lane prefetch FIFO in LDS fed by
// GLOBAL_LOAD_ASYNC_TO_LDS_B128 (ASYNCcnt), consumed via DS_LOAD_B128,
// results stored directly from VGPRs with non-temporal b128 stores.
// Keeps ~4 cachelines/lane in flight continuously with ~30 VGPRs/thread.

#define T_STEPS 16
#define DEPTH   4
#define U_TH    1.0f
#define BETA    0.5f

typedef float v4f __attribute__((ext_vector_type(4)));
typedef __attribute__((address_space(3))) v4f* lds_v4f_ptr;

__device__ __forceinline__ void async_load_b128(lds_v4f_ptr lds_dst,
                                                const float* gsrc) {
  // VDST = LDS byte address (VGPR), VADDR = 64-bit global address (VGPR pair).
  asm volatile("global_load_async_to_lds_b128 %0, %1, off th:TH_LOAD_NT"
               :: "v"(lds_dst), "v"(gsrc)
               : "memory");
}

__global__ __launch_bounds__(256) void lif_scan_async_kernel(
    const float* __restrict__ in, float* __restrict__ out, int n_v4) {
  __shared__ v4f buf[DEPTH][256];

  const int tid = threadIdx.x;
  const int v = blockIdx.x * blockDim.x + tid;
  if (v >= n_v4) return;  // grid covers n_v4 exactly; EXEC stays all-ones

  const size_t base   = (size_t)v * 4;      // this thread's 4 sites
  const size_t stride = (size_t)n_v4 * 4;   // elements per timestep (B*C*H*W)

  // Prologue: fill the LDS prefetch FIFO with timesteps 0..DEPTH-1.
#pragma unroll
  for (int k = 0; k < DEPTH; ++k) {
    async_load_b128((lds_v4f_ptr)&buf[k][tid],
                    in + (size_t)k * stride + base);
  }

  v4f u = (v4f)0.0f;
#pragma unroll
  for (int t = 0; t < T_STEPS; ++t) {
    // Async loads complete in order: waiting ASYNCcnt<=3 guarantees the
    // oldest outstanding load (timestep t) has landed in LDS.
    if (t < T_STEPS - DEPTH) {
      asm volatile("s_wait_asynccnt 0x3" ::: "memory");
    } else if (t == T_STEPS - DEPTH) {
      asm volatile("s_wait_asynccnt 0x0" ::: "memory");  // drain tail
    }

    v4f x = buf[t % DEPTH][tid];  // ds_load_b128

    // Refill this FIFO slot with timestep t+DEPTH once the ds_load is done.
    if (t + DEPTH < T_STEPS) {
      asm volatile("s_wait_dscnt 0x0" ::: "memory");
      async_load_b128((lds_v4f_ptr)&buf[t % DEPTH][tid],
                      in + (size_t)(t + DEPTH) * stride + base);
    }

    v4f s;
    u.x = __builtin_fmaf(BETA, u.x, x.x);
    u.y = __builtin_fmaf(BETA, u.y, x.y);
    u.z = __builtin_fmaf(BETA, u.z, x.z);
    u.w = __builtin_fmaf(BETA, u.w, x.w);
    s.x = (u.x > U_TH) ? 1.0f : 0.0f;
    s.y = (u.y > U_TH) ? 1.0f : 0.0f;
    s.z = (u.z > U_TH) ? 1.0f : 0.0f;
    s.w = (u.w > U_TH) ? 1.0f : 0.0f;
    u.x -= s.x * U_TH;
    u.y -= s.y * U_TH;
    u.z -= s.z * U_TH;
    u.w -= s.w * U_TH;

    __builtin_nontemporal_store(
        s, reinterpret_cast<v4f*>(out + (size_t)t * stride + base));
  }
}

extern "C" void kernel_launch(void* const* d_in, const int* in_sizes, int n_in,
                              void* d_out, int out_size, void* d_ws, size_t ws_size,
                              hipStream_t stream) {
  (void)n_in; (void)d_ws; (void)ws_size; (void)out_size;
  const float* in = (const float*)d_in[0];
  float* out = (float*)d_out;

  const int total     = in_sizes[0];            // T * B * C * H * W = 67,108,864
  const int n_spatial = total / T_STEPS;        // B * C * H * W     =  4,194,304
  const int n_v4      = n_spatial / 4;          // float4 chunks     =  1,048,576

  const int block = 256;                        // 8 waves (wave32) per block
  const int grid  = (n_v4 + block - 1) / block; // 4096 blocks, covers n_v4 exactly

  lif_scan_async_kernel<<<grid, block, 0, stream>>>(in, out, n_v4);
}